// patch2image_4801773436971
// MI455X (gfx1250) — compile-verified
//
#include <hip/hip_runtime.h>

#define PSIZE 8
#define IMG   256
#define WIN   249             // IMG + 1 - PSIZE
#define NPOS  (WIN * WIN)     // 62001
#define KDIM  (PSIZE * PSIZE) // 64

typedef float v2f __attribute__((ext_vector_type(2)));
typedef float v8f __attribute__((ext_vector_type(8)));

// ---- mask handling: inv[p] = r such that mask[r] == p, else -1 ----
__global__ void fill_inv_kernel(int* __restrict__ inv) {
    int i = blockIdx.x * blockDim.x + threadIdx.x;
    if (i < NPOS) inv[i] = -1;
}

__global__ void scatter_inv_kernel(const int* __restrict__ mask, int nmask,
                                   int* __restrict__ inv) {
    int r = blockIdx.x * blockDim.x + threadIdx.x;
    if (r < nmask) {
        int p = mask[r];
        if (p >= 0 && p < NPOS) inv[p] = r;
    }
}

// Branch-free gather of one contribution (pixel x, row y, in-patch offset k).
// Out-of-range / uncovered positions load from a safe address and are masked
// with selects, so no EXEC branching serializes the load stream. All indices
// are 32-bit (per-batch slab is 15.9 MB) so loads use SGPR-base + 32-bit
// voffset scale_offset addressing instead of 64-bit VALU address math.
template <bool USE_INV>
__device__ __forceinline__ void gather_one(const float* __restrict__ bin,
                                           const int*   __restrict__ inv,
                                           int y, int x, int k,
                                           float& a, float& ind) {
    const int pr  = y - (k >> 3);
    const int pc  = x - (k & 7);
    const bool inb = ((unsigned)pr < (unsigned)WIN) & ((unsigned)pc < (unsigned)WIN);
    const int p   = inb ? (pr * WIN + pc) : 0;          // safe index
    int rr;
    if (USE_INV) {
        rr = inv[p];                                    // unconditional load
    } else {
        rr = p;
    }
    const bool use = inb & (rr >= 0);
    const int  rrs = use ? rr : 0;                      // safe row
    const unsigned idx = (unsigned)(rrs * KDIM + k);    // 32-bit element index
    const float v  = bin[idx];                          // unconditional load
    a   = use ? v    : 0.0f;
    ind = use ? 1.0f : 0.0f;
}

// ---- fold: one wave32 -> 16 output pixels (one row y, x0..x0+15) ----
// Reduction over the 64 in-patch offsets via chained fp32 WMMA:
// D = A x ones + C  => every column of D is the row-sum of A.
template <bool USE_INV>
__global__ __launch_bounds__(256) void fold_wmma_kernel(
        const float* __restrict__ in,   // [nbatch, NPOS, KDIM]
        const int*   __restrict__ inv,  // [NPOS] (ignored if !USE_INV)
        float*       __restrict__ out,  // [nbatch, IMG, IMG]
        int nbatch) {
    const int lane = threadIdx.x & 31;
    const int wave = (blockIdx.x * blockDim.x + threadIdx.x) >> 5;

    // 16 x-tiles * 256 rows per batch = 4096 waves per batch
    const int tile = wave & 15;
    const int y    = (wave >> 4) & 255;
    const int n    = wave >> 12;
    if (n >= nbatch) return;   // wave-uniform exit

    const int x0 = tile << 4;
    const int m  = lane & 15;   // A-matrix row == pixel within tile
    const int hf = lane >> 4;   // 0: supplies K={0,1}; 1: supplies K={2,3}
    const int x  = x0 + m;

    const float* bin = in + (size_t)n * NPOS * KDIM;    // uniform SGPR base

    v8f acc = {0.f, 0.f, 0.f, 0.f, 0.f, 0.f, 0.f, 0.f};
    v8f cnt = {0.f, 0.f, 0.f, 0.f, 0.f, 0.f, 0.f, 0.f};
    const v2f ones = {1.0f, 1.0f};

#pragma unroll
    for (int t = 0; t < 16; ++t) {
        const int kbase = t * 4 + hf * 2;
        float a0, a1, i0, i1;
        gather_one<USE_INV>(bin, inv, y, x, kbase,     a0, i0);
        gather_one<USE_INV>(bin, inv, y, x, kbase + 1, a1, i1);
        const v2f a  = {a0, a1};
        const v2f ci = {i0, i1};
        // two independent accumulation chains interleave to hide WMMA latency
        acc = __builtin_amdgcn_wmma_f32_16x16x4_f32(false, a,  false, ones,
                                                    (short)0, acc, false, false);
        cnt = __builtin_amdgcn_wmma_f32_16x16x4_f32(false, ci, false, ones,
                                                    (short)0, cnt, false, false);
    }

    // Every column of D is identical; lane 0 holds sums for m=0..7 in acc[0..7],
    // lane 16 holds sums for m=8..15.
    if ((lane & 15) == 0) {
        float* orow = out + ((size_t)n * IMG + y) * IMG + x0 + hf * 8;
#pragma unroll
        for (int e = 0; e < 8; ++e) orow[e] = acc[e] / cnt[e];
    }
}

extern "C" void kernel_launch(void* const* d_in, const int* in_sizes, int n_in,
                              void* d_out, int out_size, void* d_ws, size_t ws_size,
                              hipStream_t stream) {
    const float* x    = (const float*)d_in[0];
    const int*   mask = (const int*)d_in[1];
    float*       out  = (float*)d_out;

    const int nbatch = in_sizes[0] / (NPOS * KDIM);   // 16
    const int nmask  = in_sizes[1];                   // 62001

    const int waves  = nbatch * 256 * 16;     // 65536 wave-tiles
    const int blocks = (waves + 7) / 8;       // 8 waves (256 thr) per block

    if (ws_size >= (size_t)NPOS * sizeof(int)) {
        int* inv = (int*)d_ws;
        fill_inv_kernel<<<(NPOS + 255) / 256, 256, 0, stream>>>(inv);
        scatter_inv_kernel<<<(nmask + 255) / 256, 256, 0, stream>>>(mask, nmask, inv);
        fold_wmma_kernel<true><<<blocks, 256, 0, stream>>>(x, inv, out, nbatch);
    } else {
        // identity-mask fallback (no workspace): rr == p
        fold_wmma_kernel<false><<<blocks, 256, 0, stream>>>(x, nullptr, out, nbatch);
    }

    (void)n_in; (void)out_size;
}